// SACILSTM_38371237822731
// MI455X (gfx1250) — compile-verified
//
#include <hip/hip_runtime.h>
#include <hip/hip_bf16.h>

// SACILSTM for MI455X (gfx1250): fused 4-gate WMMA-f32 LSTM.
// T=256, B=64, A=512, I=H=1024, L=2. All GEMMs via V_WMMA_F32_16X16X4_F32.

typedef float v2f __attribute__((ext_vector_type(2)));
typedef float v8f __attribute__((ext_vector_type(8)));

#define TT   256
#define BB   64
#define AA   512
#define HH   1024
#define G4   4096       // 4*H
#define BH   65536      // B*H
#define TBH  16777216   // T*B*H
#define BG   262144     // B*4H

// -----------------------------------------------------------------------------
// const[l] = a0 @ W_ah[l] + bias_ah[l] + bias_ih[l] + bias_hh[l]   -> [B, 4H]
// grid: (Ntiles=256, Mtiles=4, L=2), block: 1 wave (32 threads)
// -----------------------------------------------------------------------------
__global__ __launch_bounds__(32)
void lstm_const_kernel(const float* __restrict__ a0,     // [64,512]
                       const float* __restrict__ w_ah,   // [L,512,4096]
                       const float* __restrict__ b_ah,   // [L,4096]
                       const float* __restrict__ b_ih,
                       const float* __restrict__ b_hh,
                       float* __restrict__ cst)          // [L,64,4096]
{
    const int lane = threadIdx.x & 31;
    const int half = lane >> 4;       // 0..1
    const int lr   = lane & 15;       // 0..15
    const int nt   = blockIdx.x;      // 0..255
    const int mt   = blockIdx.y;      // 0..3
    const int l    = blockIdx.z;      // 0..1

    const float* __restrict__ W = w_ah + (size_t)l * AA * G4;
    const int row = mt * 16 + lr;     // A row (M)
    const int col = nt * 16 + lr;     // B/C col (N)

    v8f acc = {};
    for (int k = 0; k < AA; k += 4) {
        const int ka = k + 2 * half;
        v2f a = *(const v2f*)(a0 + (size_t)row * AA + ka);
        v2f b;
        b.x = W[(size_t)ka * G4 + col];
        b.y = W[(size_t)(ka + 1) * G4 + col];
        acc = __builtin_amdgcn_wmma_f32_16x16x4_f32(
                  false, a, false, b, (short)0, acc, false, false);
    }

    const float bias = b_ah[l * G4 + col] + b_ih[l * G4 + col] + b_hh[l * G4 + col];
    float* __restrict__ out = cst + (size_t)l * BG;
#pragma unroll
    for (int r = 0; r < 8; ++r) {
        const int m = mt * 16 + r + 8 * half;
        out[(size_t)m * G4 + col] = acc[r] + bias;
    }
}

// -----------------------------------------------------------------------------
// One LSTM time step for one layer:
//   gates = in @ W_ih + h_in @ W_hh + const ; fused i/f/g/o cell update.
// grid: 64 blocks (16-col slices of H), block: 128 threads = 4 waves (M tiles).
// Each wave computes 4 accumulators (one per gate) sharing the A fragment.
// -----------------------------------------------------------------------------
__global__ __launch_bounds__(128)
void lstm_step_kernel(const float* __restrict__ in,      // [64,1024] layer input at t
                      const float* __restrict__ w_ih,    // [1024,4096]
                      const float* __restrict__ w_hh,    // [1024,4096]
                      const float* __restrict__ cst,     // [64,4096]
                      const float* __restrict__ h_in,    // [64,1024] h_{t-1}
                      float* __restrict__ c_state,       // [64,1024] in-place
                      float* __restrict__ h_out,         // [64,1024]
                      float* __restrict__ hl_out,        // [64,1024] or null (t==T-1)
                      float* __restrict__ cl_out)        // [64,1024] or null
{
    const int wave = threadIdx.x >> 5;   // 0..3 -> batch tile
    const int lane = threadIdx.x & 31;
    const int half = lane >> 4;
    const int lr   = lane & 15;
    const int n0   = blockIdx.x * 16;    // column slice within H
    const int mrow = wave * 16 + lr;     // A row

    v8f acc[4] = {};                     // i, f, g, o accumulators

    // ---- K loop 1: input @ W_ih ----
    for (int k = 0; k < HH; k += 4) {
        const int ka = k + 2 * half;
        v2f a = *(const v2f*)(in + (size_t)mrow * HH + ka);
        const size_t r0 = (size_t)ka * G4;
        const size_t r1 = (size_t)(ka + 1) * G4;
#pragma unroll
        for (int g = 0; g < 4; ++g) {
            const int colg = n0 + g * HH + lr;
            v2f b;
            b.x = w_ih[r0 + colg];
            b.y = w_ih[r1 + colg];
            acc[g] = __builtin_amdgcn_wmma_f32_16x16x4_f32(
                         false, a, false, b, (short)0, acc[g], false, false);
        }
    }

    // ---- K loop 2: h_{t-1} @ W_hh ----
    for (int k = 0; k < HH; k += 4) {
        const int ka = k + 2 * half;
        v2f a = *(const v2f*)(h_in + (size_t)mrow * HH + ka);
        const size_t r0 = (size_t)ka * G4;
        const size_t r1 = (size_t)(ka + 1) * G4;
#pragma unroll
        for (int g = 0; g < 4; ++g) {
            const int colg = n0 + g * HH + lr;
            v2f b;
            b.x = w_hh[r0 + colg];
            b.y = w_hh[r1 + colg];
            acc[g] = __builtin_amdgcn_wmma_f32_16x16x4_f32(
                         false, a, false, b, (short)0, acc[g], false, false);
        }
    }

    // ---- fused LSTM cell on the accumulator layout ----
    const int col = n0 + lr;             // column within H
#pragma unroll
    for (int r = 0; r < 8; ++r) {
        const int row = wave * 16 + r + 8 * half;
        const size_t gbase = (size_t)row * G4 + col;
        const float ig = acc[0][r] + cst[gbase];
        const float fg = acc[1][r] + cst[gbase + 1 * HH];
        const float gg = acc[2][r] + cst[gbase + 2 * HH];
        const float og = acc[3][r] + cst[gbase + 3 * HH];

        const size_t hidx = (size_t)row * HH + col;
        const float cprev = c_state[hidx];

        const float si = 1.0f / (1.0f + __expf(-ig));
        const float sf = 1.0f / (1.0f + __expf(-fg));
        const float so = 1.0f / (1.0f + __expf(-og));
        const float cn = sf * cprev + si * tanhf(gg);
        const float hn = so * tanhf(cn);

        c_state[hidx] = cn;
        h_out[hidx]   = hn;
        if (hl_out) { hl_out[hidx] = hn; cl_out[hidx] = cn; }
    }
}

// -----------------------------------------------------------------------------
// Host-side orchestration (graph-capture safe: only kernel launches + async d2d)
// -----------------------------------------------------------------------------
extern "C" void kernel_launch(void* const* d_in, const int* in_sizes, int n_in,
                              void* d_out, int out_size, void* d_ws, size_t ws_size,
                              hipStream_t stream) {
    const float* attention = (const float*)d_in[0];   // [T,B,A]; only t=0 used
    const float* x         = (const float*)d_in[1];   // [T,B,I]
    const float* h0        = (const float*)d_in[3];   // [L,B,H]
    const float* c0        = (const float*)d_in[4];   // [L,B,H]
    const float* w_ah      = (const float*)d_in[5];   // [L,A,4H]
    const float* w_ih      = (const float*)d_in[6];   // [L,I,4H]
    const float* w_hh      = (const float*)d_in[7];   // [L,H,4H]
    const float* b_ah      = (const float*)d_in[8];   // [L,4H]
    const float* b_ih      = (const float*)d_in[9];
    const float* b_hh      = (const float*)d_in[10];
    float* out = (float*)d_out;  // hs [T,B,H] | h_lasts [L,B,H] | c_lasts [L,B,H]

    // Workspace layout (floats): const[2][B,4H] | h0 ping-pong[2][B,H] |
    //                            h1_init[B,H] | c_state[2][B,H]   (~3 MB total)
    float* ws     = (float*)d_ws;
    float* cst    = ws;                     // 2 * BG
    float* h0buf  = cst + 2 * BG;           // 2 * BH (layer-0 ping-pong)
    float* h1init = h0buf + 2 * BH;         // BH
    float* cstate = h1init + BH;            // 2 * BH

    // Initial states (device-to-device, capture-safe).
    hipMemcpyAsync(h0buf,  h0,      (size_t)BH * sizeof(float),     hipMemcpyDeviceToDevice, stream);
    hipMemcpyAsync(h1init, h0 + BH, (size_t)BH * sizeof(float),     hipMemcpyDeviceToDevice, stream);
    hipMemcpyAsync(cstate, c0,      (size_t)2 * BH * sizeof(float), hipMemcpyDeviceToDevice, stream);

    // Time-invariant gate bias: a0 @ W_ah + biases (both layers at once).
    lstm_const_kernel<<<dim3(256, 4, 2), 32, 0, stream>>>(
        attention, w_ah, b_ah, b_ih, b_hh, cst);

    const size_t WSTRIDE = (size_t)HH * G4;   // per-layer weight stride

    for (int t = 0; t < TT; ++t) {
        const bool last = (t == TT - 1);

        // ---- layer 0: input = x[t]; ping-pong h buffers ----
        const float* in0   = x + (size_t)t * BH;          // B*I == BH
        const float* h_in0 = h0buf + (size_t)(t & 1) * BH;
        float*       h_o0  = h0buf + (size_t)((t + 1) & 1) * BH;
        lstm_step_kernel<<<64, 128, 0, stream>>>(
            in0, w_ih, w_hh, cst, h_in0, cstate, h_o0,
            last ? out + TBH          : nullptr,
            last ? out + TBH + 2 * BH : nullptr);

        // ---- layer 1: input = layer-0 output; history lives in d_out hs ----
        const float* in1   = h_o0;
        const float* h_in1 = (t == 0) ? h1init : out + (size_t)(t - 1) * BH;
        float*       h_o1  = out + (size_t)t * BH;        // hs[t]
        lstm_step_kernel<<<64, 128, 0, stream>>>(
            in1, w_ih + WSTRIDE, w_hh + WSTRIDE, cst + BG,
            h_in1, cstate + BH, h_o1,
            last ? out + TBH + BH          : nullptr,
            last ? out + TBH + 2 * BH + BH : nullptr);
    }
}